// EntInSet_81174881894857
// MI455X (gfx1250) — compile-verified
//
#include <hip/hip_runtime.h>
#include <hip/hip_bf16.h>
#include <stdint.h>

// Problem constants from the reference (B=8192, C=4096, M=8192).
#define THREADS 256
#define WAVES   8
#define C_DIM   4096
#define M_DIM   8192
#define GROUPS  (M_DIM / (THREADS * 16))   // 2 groups of 4096 elements
                                           // thread owns 16 consecutive elems/group

#if __has_builtin(__builtin_amdgcn_global_load_async_to_lds_b128)
#define HAVE_ASYNC_LDS 1
#else
#define HAVE_ASYNC_LDS 0
#endif

// 16-byte int vector, matching the builtin's parameter type.
typedef int v4i __attribute__((vector_size(16)));
typedef __attribute__((address_space(1))) v4i* as1_v4i;
typedef __attribute__((address_space(3))) v4i* as3_v4i;

static __device__ __forceinline__ as1_v4i to_as1_v4i(const void* p) {
  return (as1_v4i)(uintptr_t)p;               // global addrs are identical in AS1
}
static __device__ __forceinline__ as3_v4i to_as3_v4i(const void* p) {
  return (as3_v4i)(uint32_t)(uintptr_t)p;     // low 32 bits of a flat LDS addr = LDS offset
}

__global__ __launch_bounds__(THREADS) void entinset_kernel(
    const float* __restrict__ x,
    const unsigned char* __restrict__ mask,
    const float* __restrict__ constr,
    float* __restrict__ out)
{
  __shared__ __attribute__((aligned(16))) float gathered[C_DIM];
#if HAVE_ASYNC_LDS
  __shared__ __attribute__((aligned(16))) float craw[M_DIM];
#endif
  __shared__ unsigned wavecnt[GROUPS][WAVES];
  __shared__ float wred[WAVES];
  __shared__ float sred[WAVES];
  __shared__ float pred[WAVES];

  const int row  = blockIdx.x;
  const int t    = threadIdx.x;
  const int lane = t & 31;
  const int w    = t >> 5;

  const float*  xrow  = x + (size_t)row * C_DIM;
  const uint4*  mrow4 = (const uint4*)(mask + (size_t)row * M_DIM);
  const float4* crow4 = (const float4*)(constr + (size_t)row * M_DIM);

#if HAVE_ASYNC_LDS
  // Stage the entire constr row in LDS via the CDNA5 async copy path.
  // Thread t stages exactly the 16B chunks it will scatter from later
  // (float4 index g*1024 + t*4 + j), so only this wave's ASYNCcnt must
  // drain — no barrier needed before the scatter reads.
  #pragma unroll
  for (int g = 0; g < GROUPS; ++g) {
    #pragma unroll
    for (int j = 0; j < 4; ++j) {
      const int ci = g * 1024 + t * 4 + j;
      __builtin_amdgcn_global_load_async_to_lds_b128(
          to_as1_v4i((const void*)(crow4 + ci)),
          to_as3_v4i((const void*)(craw + (size_t)ci * 4)),
          0, 0);
    }
  }
#endif

  // x row into registers (coalesced b128), overlapped with the async copies.
  const float4* x4 = (const float4*)xrow;
  float4 xv[4];
  #pragma unroll
  for (int k = 0; k < 4; ++k) xv[k] = x4[t + k * THREADS];

  // Mask: 16 consecutive bool bytes per thread per group = one b128 load.
  uint4 mw[GROUPS];
  #pragma unroll
  for (int g = 0; g < GROUPS; ++g) mw[g] = mrow4[g * THREADS + t];

#if HAVE_ASYNC_LDS
#if __has_builtin(__builtin_amdgcn_s_wait_asynccnt)
  __builtin_amdgcn_s_wait_asynccnt(0);
#else
  asm volatile("s_wait_asynccnt 0" ::: "memory");
#endif
#endif

  // ---- Phase 1: stable stream compaction of constr through mask ----
  // Order (g, thread, word, byte) is lexicographic == ascending element index,
  // matching stable argsort(~mask) semantics.
  unsigned base = 0;
  #pragma unroll
  for (int g = 0; g < GROUPS; ++g) {
    const uint4 m = mw[g];
    const unsigned cnt =
        (unsigned)__popc(m.x & 0x01010101u) + (unsigned)__popc(m.y & 0x01010101u) +
        (unsigned)__popc(m.z & 0x01010101u) + (unsigned)__popc(m.w & 0x01010101u);

    // Inclusive wave scan of per-thread counts (wave32).
    unsigned v = cnt;
    #pragma unroll
    for (int d = 1; d < 32; d <<= 1) {
      unsigned n = __shfl_up(v, d, 32);
      if (lane >= d) v += n;
    }
    const unsigned wtotal = __shfl(v, 31, 32);
    const unsigned excl   = v - cnt;

    if (lane == 31) wavecnt[g][w] = wtotal;
    __syncthreads();
    unsigned wpref = 0, tot = 0;
    #pragma unroll
    for (int ww = 0; ww < WAVES; ++ww) {
      const unsigned cc = wavecnt[g][ww];
      wpref += (ww < w) ? cc : 0u;
      tot   += cc;
    }

    unsigned rank = base + wpref + excl;
    base += tot;

    const unsigned words[4] = {m.x, m.y, m.z, m.w};
    #pragma unroll
    for (int j = 0; j < 4; ++j) {
      const int ci = g * 1024 + t * 4 + j;
#if HAVE_ASYNC_LDS
      const float4 v4 = ((const float4*)craw)[ci];   // ds_load_b128 from staged row
#else
      const float4 v4 = crow4[ci];                   // fallback: direct global b128
#endif
      const float vv[4] = {v4.x, v4.y, v4.z, v4.w};
      const unsigned wbits = words[j];
      #pragma unroll
      for (int b = 0; b < 4; ++b) {
        if ((wbits >> (8 * b)) & 1u) {
          if (rank < C_DIM) gathered[rank] = vv[b];
          ++rank;
        }
      }
    }
  }

  __syncthreads();   // gathered[] complete, visible to all waves

  // ---- Phase 2: fused softmax-dot ----
  float mx = -3.402823466e38f;
  #pragma unroll
  for (int k = 0; k < 4; ++k)
    mx = fmaxf(mx, fmaxf(fmaxf(xv[k].x, xv[k].y), fmaxf(xv[k].z, xv[k].w)));
  #pragma unroll
  for (int off = 16; off > 0; off >>= 1)
    mx = fmaxf(mx, __shfl_xor(mx, off, 32));
  if (lane == 0) wred[w] = mx;
  __syncthreads();
  float rmax = wred[0];
  #pragma unroll
  for (int ww = 1; ww < WAVES; ++ww) rmax = fmaxf(rmax, wred[ww]);

  const float4* g4 = (const float4*)gathered;
  float s = 0.f, p = 0.f;
  #pragma unroll
  for (int k = 0; k < 4; ++k) {
    const int i = t + k * THREADS;
    const float4 g = g4[i];
    const float e0 = __expf(xv[k].x - rmax);
    const float e1 = __expf(xv[k].y - rmax);
    const float e2 = __expf(xv[k].z - rmax);
    const float e3 = __expf(xv[k].w - rmax);
    s += (e0 + e1) + (e2 + e3);
    p += e0 * g.x + e1 * g.y + e2 * g.z + e3 * g.w;
  }
  #pragma unroll
  for (int off = 16; off > 0; off >>= 1) {
    s += __shfl_xor(s, off, 32);
    p += __shfl_xor(p, off, 32);
  }
  if (lane == 0) { sred[w] = s; pred[w] = p; }
  __syncthreads();
  if (t == 0) {
    float S = 0.f, P = 0.f;
    #pragma unroll
    for (int ww = 0; ww < WAVES; ++ww) { S += sred[ww]; P += pred[ww]; }
    float o = P / S;
    o = fminf(fmaxf(o, 0.f), 1.f);
    out[row] = o;
  }
}

extern "C" void kernel_launch(void* const* d_in, const int* in_sizes, int n_in,
                              void* d_out, int out_size, void* d_ws, size_t ws_size,
                              hipStream_t stream) {
  (void)in_sizes; (void)n_in; (void)d_ws; (void)ws_size;
  const float*         x      = (const float*)d_in[0];
  const unsigned char* mask   = (const unsigned char*)d_in[1];  // JAX bool = 1 byte
  const float*         constr = (const float*)d_in[2];
  float*               out    = (float*)d_out;
  const int B = out_size;  // 8192 rows, one block per row
  entinset_kernel<<<B, THREADS, 0, stream>>>(x, mask, constr, out);
}